// SpatialFrequencyCoRepresentation_29884382445600
// MI455X (gfx1250) — compile-verified
//
#include <hip/hip_runtime.h>

typedef float v2f __attribute__((ext_vector_type(2)));
typedef float v8f __attribute__((ext_vector_type(8)));

#define ROWS   16            // output rows per block
#define TH     (ROWS + 2)    // tile rows incl. halo
#define TW     72            // padded tile stride (66 needed)
#define WIDTH  64
#define HEIGHT 64
#define CCH    128

// 3x3 DCT matrix from the reference (_dct_mat(3)), hardcoded.
__device__ __constant__ float Gm[3][3] = {
    { 0.5773502691896258f,  0.5773502691896258f,  0.5773502691896258f },
    { 0.7071067811865476f,  0.0f,                -0.7071067811865476f },
    { 0.4082482904638630f, -0.8164965809277260f,  0.4082482904638630f }
};

__global__ void __launch_bounds__(128)
sfcr_wmma_kernel(const float* __restrict__ x,
                 const float* __restrict__ Wl,   // (4,18)
                 const float* __restrict__ bl,   // (4,)
                 float* __restrict__ out,        // (N, C*4, H, W)
                 int N)
{
    __shared__ float sW[4][12];          // folded 3x3 conv weights W_eff[o][3p+q]
    __shared__ float sT[TH * TW];        // input tile with halo

    const int tid = threadIdx.x;
    const int blk = blockIdx.x;
    const int HB  = HEIGHT / ROWS;                 // 4 row-groups
    const int h0  = (blk % HB) * ROWS;
    const int c   = (blk / HB) % CCH;
    const int n   =  blk / (HB * CCH);
    (void)N;

    // ---- fold DCT + linear into effective 3x3 weights (36 values) ----
    if (tid < 36) {
        int o = tid / 9, j = tid % 9, p = j / 3, q = j % 3;
        float acc = Wl[o * 18 + j];                // spatial part
#pragma unroll
        for (int a = 0; a < 3; ++a)
#pragma unroll
            for (int b = 0; b < 3; ++b)
                acc += Wl[o * 18 + 9 + 3 * a + b] * Gm[a][p] * Gm[b][q];
        sW[o][j] = acc;
    }

    // ---- stage (ROWS+2) x 66 tile, edge-clamped (== edge padding) ----
    const float* xin = x + (size_t)(n * CCH + c) * (HEIGHT * WIDTH);
    for (int idx = tid; idx < TH * (WIDTH + 2); idx += 128) {
        int r = idx / (WIDTH + 2), col = idx % (WIDTH + 2);
        int gh = h0 - 1 + r; gh = gh < 0 ? 0 : (gh > HEIGHT - 1 ? HEIGHT - 1 : gh);
        int gw = col - 1;    gw = gw < 0 ? 0 : (gw > WIDTH  - 1 ? WIDTH  - 1 : gw);
        sT[r * TW + col] = xin[gh * WIDTH + gw];
    }
    __syncthreads();

    const int wave = tid >> 5;
    const int lane = tid & 31;
    const int half = lane >> 4;     // which lane half (carries upper K pair)
    const int li   = lane & 15;
    const int kb   = half * 2;      // K offset within a 4-chunk for this lane half
    const int m    = li;            // A-matrix row held by this lane

    // ---- A operands: W_eff rows (row m = output o; rows 4..15 zero) ----
    // 32-bit A 16x4 layout: lane l<16 -> (M=l, K=kb+{0,1}); lane l+16 -> (M=l, K=kb+{2,3})
    v2f a0 = {0.f, 0.f}, a1 = {0.f, 0.f}, a2 = {0.f, 0.f};
    if (m < 4) {
        a0.x = sW[m][0 + kb]; a0.y = sW[m][1 + kb];      // patch pos 0..3
        a1.x = sW[m][4 + kb]; a1.y = sW[m][5 + kb];      // patch pos 4..7
        if (!half) a2.x = sW[m][8];                      // patch pos 8 (+3 zero pads)
    }

    // ---- bias folded into initial accumulator C ----
    // C/D layout: VGPR v, lanes 0-15 = (M=v, N=lane); lanes 16-31 = rows 8..15 (unused here)
    v8f cinit = {};
    if (!half) {
        float b0 = bl[0], b1 = bl[1], b2 = bl[2], b3 = bl[3];
        cinit[0] = b0; cinit[1] = b1; cinit[2] = b2; cinit[3] = b3;
    }

    float* obase = out + (size_t)((n * CCH + c) * 4) * (HEIGHT * WIDTH);

    // ---- 16 chains per wave: each = 16 pixels x 4 outputs via 3 chained WMMAs ----
    for (int it = 0; it < 16; ++it) {
        int chain = wave + 4 * it;          // 0..63, round-robin over waves
        int r  = chain >> 2;                // output row within block: 0..15
        int w0 = (chain & 3) << 4;          // pixel group start: 0,16,32,48

        // B operands: patch values for pixel (h0+r, w0+li); pos j -> sT[(r+j/3)*TW + w0+li + j%3]
        // B 4x16 layout mirrors A: VGPR selects K%2, lane half selects K/2 pair.
        const float* t = &sT[r * TW + w0 + li];
        v2f b0, b1, b2;
        b0.x = half ? t[2]            : t[0];          // pos 2 : 0
        b0.y = half ? t[TW + 0]       : t[1];          // pos 3 : 1
        b1.x = half ? t[2 * TW + 0]   : t[TW + 1];     // pos 6 : 4
        b1.y = half ? t[2 * TW + 1]   : t[TW + 2];     // pos 7 : 5
        b2.x = half ? 0.f             : t[2 * TW + 2]; // --- : pos 8
        b2.y = 0.f;

        v8f acc = cinit;
        acc = __builtin_amdgcn_wmma_f32_16x16x4_f32(false, a0, false, b0, (short)0, acc, false, false);
        acc = __builtin_amdgcn_wmma_f32_16x16x4_f32(false, a1, false, b1, (short)0, acc, false, false);
        acc = __builtin_amdgcn_wmma_f32_16x16x4_f32(false, a2, false, b2, (short)0, acc, false, false);

        // D VGPR o, lanes 0-15 = output o for pixels w0..w0+15 -> 64B contiguous stores
        if (!half) {
            float* op = obase + (size_t)(h0 + r) * WIDTH + w0 + li;
#pragma unroll
            for (int o = 0; o < 4; ++o)
                op[(size_t)o * (HEIGHT * WIDTH)] = acc[o];
        }
    }
}

extern "C" void kernel_launch(void* const* d_in, const int* in_sizes, int n_in,
                              void* d_out, int out_size, void* d_ws, size_t ws_size,
                              hipStream_t stream) {
    const float* x  = (const float*)d_in[0];   // (N, 128, 64, 64) f32
    const float* Wl = (const float*)d_in[1];   // (4, 18) f32
    const float* bl = (const float*)d_in[2];   // (4,) f32
    float* out = (float*)d_out;                // (N, 512, 64, 64) f32

    int N = in_sizes[0] / (CCH * HEIGHT * WIDTH);
    int blocks = N * CCH * (HEIGHT / ROWS);
    sfcr_wmma_kernel<<<blocks, 128, 0, stream>>>(x, Wl, bl, out, N);
}